// HashGridEncoder_41867341201410
// MI455X (gfx1250) — compile-verified
//
#include <hip/hip_runtime.h>
#include <hip/hip_bf16.h>
#include <stdint.h>

// Multi-resolution hash-grid encoder (Instant-NGP style), MI455X / gfx1250.
// L2-resident table (64 MB < 192 MB L2); bottleneck = 8 random b64 gathers
// per (point, level). wave32 mapping: level = lane&15 -> coalesced stores.

#define NLVL   16
#define THASH  (1u << 19)
#define HMASK  (THASH - 1u)
#define PTSBLK 256                 // points per block
#define NTHR   256                 // threads per block (8 waves)
#define PGRP   (NTHR / NLVL)       // 16 points in flight per iteration
#define NITER  (PTSBLK / PGRP)     // 16 iterations

typedef float        f32x2 __attribute__((ext_vector_type(2)));
typedef unsigned int u32x4 __attribute__((ext_vector_type(4)));
typedef int          i32x4 __attribute__((ext_vector_type(4)));
typedef int          i32x8 __attribute__((ext_vector_type(8)));

// RES[i] = floor(16 * (128^(1/15))^i)
__constant__ int c_res[NLVL] = {16, 22, 30, 42, 58, 80, 111, 153,
                                212, 294, 406, 561, 776, 1072, 1482, 2048};

__global__ __launch_bounds__(NTHR)
void hashgrid_fwd(const float* __restrict__ xyz,
                  const float* __restrict__ tables,
                  float* __restrict__ out, int npts)
{
    __shared__ float sxyz[PTSBLK * 3];
    const int tid   = threadIdx.x;
    const int level = tid & (NLVL - 1);
    const int base  = blockIdx.x * PTSBLK;

    // ---- Stage this block's 256 points (3 KB) into LDS via the Tensor Data
    // ---- Mover (CDNA5 TDM). One wave issues the DMA descriptor; it waits on
    // ---- TENSORcnt, then the workgroup barrier releases everyone.
#if __has_builtin(__builtin_amdgcn_tensor_load_to_lds) && __has_builtin(__builtin_amdgcn_s_wait_tensorcnt)
    if (tid < 32) {
        const unsigned long long ga =
            (unsigned long long)(uintptr_t)(xyz + (size_t)base * 3);
        const unsigned la = (unsigned)(uintptr_t)&sxyz[0]; // addr[31:0] == LDS byte offset
        long long rem3 = (long long)(npts - base) * 3;
        unsigned td0 = rem3 >= (long long)(PTSBLK * 3) ? (unsigned)(PTSBLK * 3)
                                                       : (unsigned)rem3;
        // D# group 0 (ISA 8.3): count=1, lds_addr, global_addr[56:0], type=2
        u32x4 g0;
        g0.x = 1u;
        g0.y = la;
        g0.z = (unsigned)ga;
        g0.w = ((unsigned)(ga >> 32) & 0x01FFFFFFu) | 0x80000000u;
        // D# group 1 (ISA 8.4): data_size=2 (4B), 1-D tile of 768 elements
        i32x8 g1;
        g1[0] = (int)(2u << 16);                    // data_size=4B, no pad/iter/multicast
        g1[1] = (int)((td0 & 0xFFFFu) << 16);       // tensor_dim0[15:0]
        g1[2] = (int)((td0 >> 16) | (1u << 16));    // tensor_dim0[31:16] | tensor_dim1=1
        g1[3] = (int)((unsigned)(PTSBLK * 3) << 16);// tile_dim0 = 768
        g1[4] = 0;                                  // tile_dim1=0 (unused), tile_dim2=0
        g1[5] = (int)td0;                           // tensor_dim0_stride[31:0]
        g1[6] = 0;
        g1[7] = 0;
        i32x4 gz4 = (i32x4){0, 0, 0, 0};            // groups 2/3 unused for <=2D
        i32x8 gz8 = (i32x8){0, 0, 0, 0, 0, 0, 0, 0};
        __builtin_amdgcn_tensor_load_to_lds(g0, g1, gz4, gz4, gz8, 0);
        __builtin_amdgcn_s_wait_tensorcnt(0);
    }
    __syncthreads();
#else
    for (int i = tid; i < PTSBLK * 3; i += NTHR) {
        long long gi = (long long)base * 3 + i;
        sxyz[i] = (gi < (long long)npts * 3) ? xyz[gi] : 0.0f;
    }
    __syncthreads();
#endif

    const float cell = 2.0f / (float)c_res[level];
    const f32x2* __restrict__ tb = (const f32x2*)tables + (size_t)level * THASH;
    f32x2* __restrict__ op = (f32x2*)out;

    for (int it = 0; it < NITER; ++it) {
        const int pl = (tid >> 4) + it * PGRP;   // local point (broadcast per 16 lanes)
        const int gp = base + pl;
        const float x = sxyz[pl * 3 + 0];
        const float y = sxyz[pl * 3 + 1];
        const float z = sxyz[pl * 3 + 2];

        // min_index = floor((p - BB_MIN)/cell); d = (p - (min_index*cell + BB_MIN))/cell
        const float fx = floorf((x + 1.0f) / cell);
        const float fy = floorf((y + 1.0f) / cell);
        const float fz = floorf((z + 1.0f) / cell);
        const float dx = (x - (fx * cell - 1.0f)) / cell;
        const float dy = (y - (fy * cell - 1.0f)) / cell;
        const float dz = (z - (fz * cell - 1.0f)) / cell;

        const unsigned ix = (unsigned)(int)fx;
        const unsigned iy = (unsigned)(int)fy;
        const unsigned iz = (unsigned)(int)fz;
        // hash = x ^ y*2654435761 ^ z*805459861 mod 2^19 ; (c+1)*K == c*K + K
        const unsigned hy0 = iy * 2654435761u, hy1 = hy0 + 2654435761u;
        const unsigned hz0 = iz * 805459861u,  hz1 = hz0 + 805459861u;
        const unsigned ix1 = ix + 1u;

        // 8 independent global_load_b64 gathers (L2-resident table)
        const f32x2 e000 = tb[(ix  ^ hy0 ^ hz0) & HMASK];
        const f32x2 e001 = tb[(ix  ^ hy0 ^ hz1) & HMASK];
        const f32x2 e010 = tb[(ix  ^ hy1 ^ hz0) & HMASK];
        const f32x2 e011 = tb[(ix  ^ hy1 ^ hz1) & HMASK];
        const f32x2 e100 = tb[(ix1 ^ hy0 ^ hz0) & HMASK];
        const f32x2 e101 = tb[(ix1 ^ hy0 ^ hz1) & HMASK];
        const f32x2 e110 = tb[(ix1 ^ hy1 ^ hz0) & HMASK];
        const f32x2 e111 = tb[(ix1 ^ hy1 ^ hz1) & HMASK];

        const float ux = 1.0f - dx, uy = 1.0f - dy, uz = 1.0f - dz;
        const float w00 = ux * uy, w01 = ux * dy, w10 = dx * uy, w11 = dx * dy;

        f32x2 acc = (w00 * uz) * e000;
        acc += (w00 * dz) * e001;
        acc += (w01 * uz) * e010;
        acc += (w01 * dz) * e011;
        acc += (w10 * uz) * e100;
        acc += (w10 * dz) * e101;
        acc += (w11 * uz) * e110;
        acc += (w11 * dz) * e111;

        // out[p, level*2 : level*2+2] ; address == 8*(base*16 + it*256 + tid)
        // -> perfectly coalesced. NT store: don't let the 128 MB write-once
        // output evict the 64 MB L2-resident hash tables.
        if (gp < npts)
            __builtin_nontemporal_store(acc, op + (size_t)gp * NLVL + level);
    }
}

extern "C" void kernel_launch(void* const* d_in, const int* in_sizes, int n_in,
                              void* d_out, int out_size, void* d_ws, size_t ws_size,
                              hipStream_t stream) {
    const float* xyz    = (const float*)d_in[0];   // [N, 3] f32
    const float* tables = (const float*)d_in[1];   // [16, 2^19, 2] f32
    float* out          = (float*)d_out;           // [N, 32] f32
    const int npts = in_sizes[0] / 3;
    const int blocks = (npts + PTSBLK - 1) / PTSBLK;
    hashgrid_fwd<<<blocks, NTHR, 0, stream>>>(xyz, tables, out, npts);
}